// AssignmentDecoder_24283745092069
// MI455X (gfx1250) — compile-verified
//
#include <hip/hip_runtime.h>
#include <math.h>

#define Bsz 32
#define Nrb 256
#define Mtk 256
#define Dd  512

typedef __attribute__((ext_vector_type(2))) float v2f;
typedef __attribute__((ext_vector_type(8))) float v8f;

#define FINF __builtin_inff()

// ---------------------------------------------------------------------------
// fp32 WMMA: D(16x16) = A(16x4) * B(4x16) + C   (V_WMMA_F32_16X16X4_F32)
// ---------------------------------------------------------------------------
__device__ __forceinline__ v8f wmma_f32(v2f a, v2f b, v8f c) {
  // 8 args: (neg_a, A, neg_b, B, c_mod, C, reuse_a, reuse_b)
  return __builtin_amdgcn_wmma_f32_16x16x4_f32(false, a, false, b, (short)0, c,
                                               false, false);
}

// ---------------------------------------------------------------------------
// Bias dots: a_r = h_robots . v[:D], a_t = h_tasks . v[D:]. One wave per row.
// ---------------------------------------------------------------------------
__global__ void __launch_bounds__(256)
bias_kernel(const float* __restrict__ hr, const float* __restrict__ ht,
            const float* __restrict__ v, float* __restrict__ aR,
            float* __restrict__ aT) {
  const int gw   = (blockIdx.x * blockDim.x + threadIdx.x) >> 5;
  const int lane = threadIdx.x & 31;
  if (gw >= Bsz * (Nrb + Mtk)) return;
  const float* row;
  const float* vv;
  float* out;
  int idx;
  if (gw < Bsz * Nrb) {
    idx = gw;            row = hr + (size_t)idx * Dd; vv = v;      out = aR;
  } else {
    idx = gw - Bsz * Nrb; row = ht + (size_t)idx * Dd; vv = v + Dd; out = aT;
  }
  float s = 0.f;
  for (int d = lane; d < Dd; d += 32) s += row[d] * vv[d];
#pragma unroll
  for (int m = 16; m >= 1; m >>= 1) s += __shfl_xor(s, m, 32);
  if (lane == 0) out[idx] = s;
}

// ---------------------------------------------------------------------------
// GEMM1: G[b] (256 x 512) = h_tasks[b] (256 x 512) @ W (512 x 512)
// Block = 64x64 tile (4 waves, each 32x32 = 2x2 wmma tiles). K-step 4.
// ---------------------------------------------------------------------------
__global__ void __launch_bounds__(128)
gemm1_kernel(const float* __restrict__ Ht, const float* __restrict__ Wm,
             float* __restrict__ G) {
  const int b    = blockIdx.y;
  const int tr   = blockIdx.x >> 3;   // m-tile of 64 (0..3)
  const int tc   = blockIdx.x & 7;    // d-tile of 64 (0..7)
  const int w    = threadIdx.x >> 5;
  const int lane = threadIdx.x & 31;
  const int lr   = lane & 15;
  const int hf   = lane >> 4;
  const int r0   = tr * 64 + (w >> 1) * 32;
  const int c0   = tc * 64 + (w & 1) * 32;
  const float* A = Ht + (size_t)b * Mtk * Dd;
  float* C       = G + (size_t)b * Mtk * Dd;

  v8f acc00 = {}, acc01 = {}, acc10 = {}, acc11 = {};
  for (int k = 0; k < Dd; k += 4) {
    const int ks = k + 2 * hf;
    // A fragment (16x4 layout): lanes 0-15 -> K=k,k+1 ; lanes 16-31 -> k+2,k+3
    v2f a0 = *reinterpret_cast<const v2f*>(A + (size_t)(r0 + lr) * Dd + ks);
    v2f a1 = *reinterpret_cast<const v2f*>(A + (size_t)(r0 + 16 + lr) * Dd + ks);
    // B fragment: B[k][n] = W[k][n], N across lanes, K rows per VGPR
    v2f b0, b1;
    b0.x = Wm[(size_t)ks * Dd + c0 + lr];
    b0.y = Wm[(size_t)(ks + 1) * Dd + c0 + lr];
    b1.x = Wm[(size_t)ks * Dd + c0 + 16 + lr];
    b1.y = Wm[(size_t)(ks + 1) * Dd + c0 + 16 + lr];
    acc00 = wmma_f32(a0, b0, acc00);
    acc01 = wmma_f32(a0, b1, acc01);
    acc10 = wmma_f32(a1, b0, acc10);
    acc11 = wmma_f32(a1, b1, acc11);
  }
#pragma unroll
  for (int i = 0; i < 8; i++) {
    const int rA = r0 + i + 8 * hf;        // C/D layout: VGPR i -> row i / i+8
    const int rB = rA + 16;
    C[(size_t)rA * Dd + c0 + lr]      = acc00[i];
    C[(size_t)rA * Dd + c0 + 16 + lr] = acc01[i];
    C[(size_t)rB * Dd + c0 + lr]      = acc10[i];
    C[(size_t)rB * Dd + c0 + 16 + lr] = acc11[i];
  }
}

// ---------------------------------------------------------------------------
// GEMM2 (NT): S[b][m][n] = (G[b,m,:] . h_r[b,n,:] + aT[m] + aR[n]) / sqrt(D)
// Both operands K-contiguous -> both fragments are float2 row loads.
// Output stored transposed (b,m,n) so decode reads coalesced rows.
// ---------------------------------------------------------------------------
__global__ void __launch_bounds__(128)
gemm2_kernel(const float* __restrict__ G, const float* __restrict__ Hr,
             const float* __restrict__ aR, const float* __restrict__ aT,
             float* __restrict__ S) {
  const int b    = blockIdx.y;
  const int tr   = blockIdx.x >> 2;   // m-tile of 64 (0..3)
  const int tc   = blockIdx.x & 3;    // n-tile of 64 (0..3)
  const int w    = threadIdx.x >> 5;
  const int lane = threadIdx.x & 31;
  const int lr   = lane & 15;
  const int hf   = lane >> 4;
  const int r0   = tr * 64 + (w >> 1) * 32;   // m
  const int c0   = tc * 64 + (w & 1) * 32;    // n
  const float* A  = G + (size_t)b * Mtk * Dd;
  const float* Bm = Hr + (size_t)b * Nrb * Dd;

  v8f acc00 = {}, acc01 = {}, acc10 = {}, acc11 = {};
  for (int k = 0; k < Dd; k += 4) {
    const int ks = k + 2 * hf;
    v2f a0 = *reinterpret_cast<const v2f*>(A + (size_t)(r0 + lr) * Dd + ks);
    v2f a1 = *reinterpret_cast<const v2f*>(A + (size_t)(r0 + 16 + lr) * Dd + ks);
    // B[k][n] = Hr[n][k] -> contiguous along k in Hr row n
    v2f b0 = *reinterpret_cast<const v2f*>(Bm + (size_t)(c0 + lr) * Dd + ks);
    v2f b1 = *reinterpret_cast<const v2f*>(Bm + (size_t)(c0 + 16 + lr) * Dd + ks);
    acc00 = wmma_f32(a0, b0, acc00);
    acc01 = wmma_f32(a0, b1, acc01);
    acc10 = wmma_f32(a1, b0, acc10);
    acc11 = wmma_f32(a1, b1, acc11);
  }
  const float inv = 0.04419417382415922f;  // 1/sqrt(512)
#pragma unroll
  for (int i = 0; i < 8; i++) {
    const int rA = r0 + i + 8 * hf;
    const int rB = rA + 16;
    const int cA = c0 + lr, cB = c0 + 16 + lr;
    S[((size_t)b * Mtk + rA) * Nrb + cA] =
        (acc00[i] + aT[b * Mtk + rA] + aR[b * Nrb + cA]) * inv;
    S[((size_t)b * Mtk + rA) * Nrb + cB] =
        (acc01[i] + aT[b * Mtk + rA] + aR[b * Nrb + cB]) * inv;
    S[((size_t)b * Mtk + rB) * Nrb + cA] =
        (acc10[i] + aT[b * Mtk + rB] + aR[b * Nrb + cA]) * inv;
    S[((size_t)b * Mtk + rB) * Nrb + cB] =
        (acc11[i] + aT[b * Mtk + rB] + aR[b * Nrb + cB]) * inv;
  }
}

// ---------------------------------------------------------------------------
// Stable argsort of -priority per batch: rank counting (O(M^2), M=256).
// ---------------------------------------------------------------------------
__global__ void __launch_bounds__(256)
sort_kernel(const float* __restrict__ prio, int* __restrict__ order) {
  const int b = blockIdx.x;
  const int i = threadIdx.x;
  __shared__ float p[Mtk];
  p[i] = prio[b * Mtk + i];
  __syncthreads();
  const float pi = p[i];
  int rank = 0;
  for (int k = 0; k < Mtk; k++) {
    const float pk = p[k];
    rank += (pk > pi) || (pk == pi && k < i);
  }
  order[b * Mtk + rank] = i;
}

// ---------------------------------------------------------------------------
// Block reductions (wave32; 8 waves of 256 threads).
// ---------------------------------------------------------------------------
__device__ __forceinline__ float blockMax(float v, float* sm) {
  const int lane = threadIdx.x & 31, wid = threadIdx.x >> 5;
#pragma unroll
  for (int m = 16; m >= 1; m >>= 1) v = fmaxf(v, __shfl_xor(v, m, 32));
  if (lane == 0) sm[wid] = v;
  __syncthreads();
  if (wid == 0) {
    float x = (lane < 8) ? sm[lane] : -FINF;
#pragma unroll
    for (int m = 4; m >= 1; m >>= 1) x = fmaxf(x, __shfl_xor(x, m, 32));
    if (lane == 0) sm[0] = x;
  }
  __syncthreads();
  const float r = sm[0];
  __syncthreads();
  return r;
}

__device__ __forceinline__ float blockSum(float v, float* sm) {
  const int lane = threadIdx.x & 31, wid = threadIdx.x >> 5;
#pragma unroll
  for (int m = 16; m >= 1; m >>= 1) v += __shfl_xor(v, m, 32);
  if (lane == 0) sm[wid] = v;
  __syncthreads();
  if (wid == 0) {
    float x = (lane < 8) ? sm[lane] : 0.f;
#pragma unroll
    for (int m = 4; m >= 1; m >>= 1) x += __shfl_xor(x, m, 32);
    if (lane == 0) sm[0] = x;
  }
  __syncthreads();
  const float r = sm[0];
  __syncthreads();
  return r;
}

__device__ __forceinline__ int blockMinI(int v, int* sm) {
  const int lane = threadIdx.x & 31, wid = threadIdx.x >> 5;
#pragma unroll
  for (int m = 16; m >= 1; m >>= 1) v = min(v, __shfl_xor(v, m, 32));
  if (lane == 0) sm[wid] = v;
  __syncthreads();
  if (wid == 0) {
    int x = (lane < 8) ? sm[lane] : 0x7fffffff;
#pragma unroll
    for (int m = 4; m >= 1; m >>= 1) x = min(x, __shfl_xor(x, m, 32));
    if (lane == 0) sm[0] = x;
  }
  __syncthreads();
  const int r = sm[0];
  __syncthreads();
  return r;
}

// ---------------------------------------------------------------------------
// Sequential greedy decode: one block per batch, thread n = robot n.
// ---------------------------------------------------------------------------
__global__ void __launch_bounds__(256)
decode_kernel(const float* __restrict__ S, const int* __restrict__ order,
              const float* __restrict__ robot_cap,
              const float* __restrict__ robot_battery,
              const float* __restrict__ robot_pos,
              const float* __restrict__ robot_erate,
              const float* __restrict__ task_weight,
              const float* __restrict__ task_pick,
              const float* __restrict__ task_drop,
              int* __restrict__ assign_out, float* __restrict__ logp_out) {
  const int b = blockIdx.x;
  const int n = threadIdx.x;
  __shared__ float smf[8];
  __shared__ int smi[8];

  const float cap = robot_cap[b * Nrb + n];
  const float er  = robot_erate[b * Nrb + n];
  float bat = robot_battery[b * Nrb + n];
  float px  = robot_pos[(b * Nrb + n) * 2 + 0];
  float py  = robot_pos[(b * Nrb + n) * 2 + 1];

  for (int j = 0; j < Mtk; j++) {
    const int t    = order[b * Mtk + j];
    const float tw = task_weight[b * Mtk + t];
    const float pk0 = task_pick[(b * Mtk + t) * 2 + 0];
    const float pk1 = task_pick[(b * Mtk + t) * 2 + 1];
    const float dp0 = task_drop[(b * Mtk + t) * 2 + 0];
    const float dp1 = task_drop[(b * Mtk + t) * 2 + 1];
    const float s   = S[((size_t)b * Mtk + t) * Nrb + n];  // coalesced row

    const float dx = px - pk0, dy = py - pk1;
    const float ex = pk0 - dp0, ey = pk1 - dp1;
    const float dtot = sqrtf(dx * dx + dy * dy) + sqrtf(ex * ex + ey * ey);
    const bool infeas = (cap < tw) || (dtot * er > bat);

    float logit = infeas ? -FINF : s;
    float mx = blockMax(logit, smf);
    const bool all_masked = !(mx > -FINF);
    if (all_masked) { logit = 0.0f; mx = 0.0f; }

    const float e = expf(logit - mx);  // exp(-inf)=0 for masked lanes
    const float sum = blockSum(e, smf);

    const int cand = (logit == mx) ? n : 0x7fffffff;  // first max wins (argmax)
    const int chosen = blockMinI(cand, smi);

    const float log_p = logf(1.0f / sum + 1e-8f);  // probs[chosen] = 1/sumexp

    if (n == 0) {
      assign_out[b * Mtk + t] = chosen;   // indexed by task id
      logp_out[b * Mtk + j]   = log_p;    // indexed by scan step
    }
    if (n == chosen) {
      bat -= dtot * er;
      px = dp0;
      py = dp1;
    }
    __syncthreads();
  }
}

// ---------------------------------------------------------------------------
extern "C" void kernel_launch(void* const* d_in, const int* in_sizes, int n_in,
                              void* d_out, int out_size, void* d_ws,
                              size_t ws_size, hipStream_t stream) {
  const float* h_robots      = (const float*)d_in[0];
  const float* h_tasks       = (const float*)d_in[1];
  const float* robot_cap     = (const float*)d_in[2];
  const float* robot_battery = (const float*)d_in[3];
  const float* robot_pos     = (const float*)d_in[4];
  const float* robot_erate   = (const float*)d_in[5];
  const float* task_weight   = (const float*)d_in[6];
  const float* task_pick     = (const float*)d_in[7];
  const float* task_drop     = (const float*)d_in[8];
  const float* task_priority = (const float*)d_in[9];
  const float* W_a_w         = (const float*)d_in[10];
  const float* v_a_w         = (const float*)d_in[11];

  float* ws  = (float*)d_ws;
  float* aR  = ws;                               // B*N
  float* aT  = aR + Bsz * Nrb;                   // B*M
  float* G   = aT + Bsz * Mtk;                   // B*M*D
  float* S   = G + (size_t)Bsz * Mtk * Dd;       // B*M*N (scores transposed)
  int* order = (int*)(S + (size_t)Bsz * Mtk * Nrb);  // B*M

  int* assign_out = (int*)d_out;                 // (B,M) int32
  float* logp_out = (float*)d_out + Bsz * Mtk;   // (B,M) f32

  bias_kernel<<<(Bsz * (Nrb + Mtk)) / 8, 256, 0, stream>>>(h_robots, h_tasks,
                                                           v_a_w, aR, aT);
  gemm1_kernel<<<dim3(32, Bsz), 128, 0, stream>>>(h_tasks, W_a_w, G);
  gemm2_kernel<<<dim3(16, Bsz), 128, 0, stream>>>(G, h_robots, aR, aT, S);
  sort_kernel<<<Bsz, Mtk, 0, stream>>>(task_priority, order);
  decode_kernel<<<Bsz, Nrb, 0, stream>>>(S, order, robot_cap, robot_battery,
                                         robot_pos, robot_erate, task_weight,
                                         task_pick, task_drop, assign_out,
                                         logp_out);
}